// RNadeOutputLayer_65790309040623
// MI455X (gfx1250) — compile-verified
//
#include <hip/hip_runtime.h>
#include <math.h>

typedef __attribute__((ext_vector_type(2))) float v2f;
typedef __attribute__((ext_vector_type(8))) float v8f;

#define B_ 256
#define K_ 16
#define H_ 512
#define D_ 64
#define HALF_LOG_2PI 0.9189385332046727f

// LDS accumulator: [mat][b'][k][d'] with d-dim padded 8->9 to spread banks
#define AIDX(m, bp, k, dd) ((((m) * 16 + (bp)) * 16 + (k)) * 9 + (dd))

__global__ __launch_bounds__(256) void rnade_gemm_epilogue(
    const float* __restrict__ x, const float* __restrict__ input_var,
    const float* __restrict__ Va, const float* __restrict__ Vm,
    const float* __restrict__ Vs, const float* __restrict__ ba,
    const float* __restrict__ bm, const float* __restrict__ bs,
    float* __restrict__ ws) {
  __shared__ float acc_lds[3 * 16 * 16 * 9];  // ~27.6 KB of 320 KB

  const int tid = threadIdx.x;
  const int wave = tid >> 5;
  const int lane = tid & 31;
  const int b0 = (blockIdx.x & 15) * 16;  // 16 b rows per block
  const int d0 = (blockIdx.x >> 4) * 8;   // 8 d per block

  // zero LDS partial-sum buffer
  for (int i = tid; i < 3 * 16 * 16 * 9; i += 256) acc_lds[i] = 0.0f;
  __syncthreads();

  const int dq = wave & 1;          // which d-quad (0..1) -> 4 d's
  const int hq = wave >> 1;         // h quarter (0..3) -> 128 h's
  const int dbase = d0 + dq * 4;
  const int row = lane & 15;        // b' for A-frag, k for B-frag
  const int hk2 = (lane >> 4) * 2;  // per-half h sub-offset (0 or 2)

  v8f acc[3][4];
#pragma unroll
  for (int m = 0; m < 3; ++m)
#pragma unroll
    for (int d = 0; d < 4; ++d)
#pragma unroll
      for (int c = 0; c < 8; ++c) acc[m][d][c] = 0.0f;

  const int h_begin = hq * 128;
  for (int h = h_begin; h < h_begin + 128; h += 4) {
    const int hh = h + hk2;
    // A fragments: x[b0+row, hh(+1), dbase..dbase+3] -- 16B coalesced in d
    const float4 a0 = *(const float4*)(x + ((size_t)(b0 + row) * H_ + hh) * D_ + dbase);
    const float4 a1 = *(const float4*)(x + ((size_t)(b0 + row) * H_ + hh + 1) * D_ + dbase);
    // B fragments: V[row(=k), hh(+1), dbase..dbase+3] for the 3 matrices
    const float4 p0 = *(const float4*)(Va + ((size_t)row * H_ + hh) * D_ + dbase);
    const float4 p1 = *(const float4*)(Va + ((size_t)row * H_ + hh + 1) * D_ + dbase);
    const float4 q0 = *(const float4*)(Vm + ((size_t)row * H_ + hh) * D_ + dbase);
    const float4 q1 = *(const float4*)(Vm + ((size_t)row * H_ + hh + 1) * D_ + dbase);
    const float4 r0 = *(const float4*)(Vs + ((size_t)row * H_ + hh) * D_ + dbase);
    const float4 r1 = *(const float4*)(Vs + ((size_t)row * H_ + hh + 1) * D_ + dbase);

    const float* a0p = (const float*)&a0;
    const float* a1p = (const float*)&a1;
    const float* bp0[3] = {(const float*)&p0, (const float*)&q0, (const float*)&r0};
    const float* bp1[3] = {(const float*)&p1, (const float*)&q1, (const float*)&r1};

#pragma unroll
    for (int d = 0; d < 4; ++d) {
      v2f af;
      af.x = a0p[d];
      af.y = a1p[d];
#pragma unroll
      for (int m = 0; m < 3; ++m) {
        v2f bf;
        bf.x = bp0[m][d];
        bf.y = bp1[m][d];
        // D(16x16 f32) = A(16x4 f32) x B(4x16 f32) + C
        acc[m][d] = __builtin_amdgcn_wmma_f32_16x16x4_f32(
            false, af, false, bf, (short)0, acc[m][d], false, false);
      }
    }
  }

  // combine the 4 h-quarter partials per (b-tile, d-quad) via LDS f32 atomics
  const int bhalf = (lane >> 4) << 3;  // +8 for upper-lane half
#pragma unroll
  for (int i = 0; i < 8; ++i) {
    const int bp = i + bhalf;  // M index of C element i for this lane
    const int k = row;         // N index = lane & 15
#pragma unroll
    for (int m = 0; m < 3; ++m)
#pragma unroll
      for (int d = 0; d < 4; ++d)
        atomicAdd(&acc_lds[AIDX(m, bp, k, dq * 4 + d)], acc[m][d][i]);
  }
  __syncthreads();

  // Epilogue: one thread per (b', d') pair; 128 active threads
  if (tid < 128) {
    const int bp = tid >> 3;
    const int dd = tid & 7;
    const int gd = d0 + dd;
    const float y = input_var[(size_t)(b0 + bp) * D_ + gd];

    float ea[16], g[16];
    float maxa = -INFINITY, maxg = -INFINITY;
#pragma unroll
    for (int k = 0; k < 16; ++k) {
      const float A = acc_lds[AIDX(0, bp, k, dd)] + ba[k * D_ + gd];
      const float M = acc_lds[AIDX(1, bp, k, dd)] + bm[k * D_ + gd];
      float S = acc_lds[AIDX(2, bp, k, dd)] + bs[k * D_ + gd];
      S = fminf(fmaxf(S, -20.0f), 20.0f);
      const float sig = __expf(S);
      const float z = (y - M) / sig;
      ea[k] = A;
      g[k] = -0.5f * z * z + A - S;
      maxa = fmaxf(maxa, A);
      maxg = fmaxf(maxg, g[k]);
    }
    float sa = 0.0f, sg = 0.0f;
#pragma unroll
    for (int k = 0; k < 16; ++k) {
      sa += __expf(ea[k] - maxa);
      sg += __expf(g[k] - maxg);
    }
    // r = LSE_k(log_p) = LSE(g) - LSE(ea) - 0.5*log(2*pi)
    const float r = (maxg + __logf(sg)) - (maxa + __logf(sa)) - HALF_LOG_2PI;
    ws[(size_t)(b0 + bp) * D_ + gd] = r;
  }
}

// Deterministic reduction over d (avoids fp global atomics)
__global__ __launch_bounds__(256) void rnade_reduce(const float* __restrict__ ws,
                                                    float* __restrict__ out) {
  const int b = blockIdx.x * blockDim.x + threadIdx.x;
  if (b < B_) {
    float s = 0.0f;
    const float* p = ws + (size_t)b * D_;
#pragma unroll 8
    for (int d = 0; d < D_; ++d) s += p[d];
    out[b] = s;
  }
}

extern "C" void kernel_launch(void* const* d_in, const int* in_sizes, int n_in,
                              void* d_out, int out_size, void* d_ws, size_t ws_size,
                              hipStream_t stream) {
  const float* x = (const float*)d_in[0];
  const float* input_var = (const float*)d_in[1];
  const float* Va = (const float*)d_in[2];
  const float* Vm = (const float*)d_in[3];
  const float* Vs = (const float*)d_in[4];
  const float* ba = (const float*)d_in[5];
  const float* bm = (const float*)d_in[6];
  const float* bs = (const float*)d_in[7];
  float* out = (float*)d_out;
  float* ws = (float*)d_ws;  // 256*64 floats = 64 KB

  // grid: 16 b-tiles x 8 d-groups = 128 blocks, 8 waves each
  rnade_gemm_epilogue<<<dim3(128), dim3(256), 0, stream>>>(
      x, input_var, Va, Vm, Vs, ba, bm, bs, ws);
  rnade_reduce<<<dim3(1), dim3(256), 0, stream>>>(ws, out);
}